// GCN_74397423501922
// MI455X (gfx1250) — compile-verified
//
#include <hip/hip_runtime.h>

#define USE_ASYNC_LDS 1

// Problem constants (match reference)
#define BGR   128                 // graphs
#define NSN   200                 // nodes per graph
#define FIN   200                 // input features
#define HID   128                 // hidden
#define DEG   64                  // in-edges per node
#define NN    (BGR * NSN)         // 25600 nodes
#define EE    ((size_t)NN * DEG)  // 1,638,400 edges

#define KC    64                  // K chunk staged in LDS (32 KB of fp32 x 128)
#define NC    128                 // GEMM output width (all GEMMs here have Nc=128)
#define LIN_KSPAN  256            // K slice per block for the lincomb K-split
#define LIN_CHUNKS (NSN * HID / LIN_KSPAN)   // 25600/256 = 100

typedef __attribute__((ext_vector_type(2))) float v2f;
typedef __attribute__((ext_vector_type(8))) float v8f;

// ---------------------------------------------------------------------------
// 1) Degree + dinv: dst blocks are contiguous (node n owns edges [n*64,(n+1)*64))
//    -> one wave per node, shuffle reduction, no atomics.
// ---------------------------------------------------------------------------
__global__ void deg_dinv_kernel(const float* __restrict__ ew, float* __restrict__ dinv) {
    int wave = blockIdx.x * (blockDim.x >> 5) + (threadIdx.x >> 5);
    int lane = threadIdx.x & 31;
    if (wave >= NN) return;
    const float* e = ew + (size_t)wave * DEG;
    float s = fmaxf(e[lane], 0.0f) + fmaxf(e[lane + 32], 0.0f);
    #pragma unroll
    for (int off = 16; off > 0; off >>= 1) s += __shfl_down(s, off);
    if (lane == 0) dinv[wave] = rsqrtf(s + 1.0f);   // deg >= 1 always
}

// ---------------------------------------------------------------------------
// 2) Edge normalization, computed once and reused for all 3 conv layers.
// ---------------------------------------------------------------------------
__global__ void norm_kernel(const int* __restrict__ src, const int* __restrict__ dst,
                            const float* __restrict__ ew, const float* __restrict__ dinv,
                            float* __restrict__ nrm) {
    size_t e = (size_t)blockIdx.x * blockDim.x + threadIdx.x;
    if (e >= EE) return;
    float w = ew[e];
    w = w > 0.0f ? w : 0.0f;
    nrm[e] = dinv[src[e]] * w * dinv[dst[e]];
}

// ---------------------------------------------------------------------------
// 3) Strip GEMM via V_WMMA_F32_16X16X4_F32.
//    Block = 256 threads (8 waves). Wave w owns M-tile (blockIdx.x*8+w) and the
//    full 128-wide N extent: 8 f32 accumulators (64 VGPRs). W is staged in LDS
//    in KC=64-row chunks (async copy -> no VGPR round trip); per k-step one b64
//    A load feeds 8 WMMAs reading B frags from LDS (ds_load_2addr-fusable).
//    K-split support: grid.y selects slice [y*kSpan, y*kSpan+kSpan); partials
//    go to C + y*partialStride (partialStride=0 => direct output).
// ---------------------------------------------------------------------------
__global__ void gemm_strip_wmma(const float* __restrict__ A, const float* __restrict__ W,
                                float* __restrict__ C, const float* __restrict__ bias,
                                int K, int kSpan, size_t partialStride) {
    __shared__ float s_w[KC * NC];          // 32 KB
    const int tid  = threadIdx.x;
    const int wave = tid >> 5;
    const int lane = tid & 31;
    const int half = lane >> 4;             // 0: K pair {0,1}; 1: K pair {2,3}
    const int r    = lane & 15;

    const int tm    = blockIdx.x * 8 + wave;
    const int m     = tm * 16 + r;
    const int kBase = blockIdx.y * kSpan;
    const int kEnd  = kBase + kSpan;        // caller guarantees <= K

    float* Cw = C + (size_t)blockIdx.y * partialStride;

    v8f acc[8];
    #pragma unroll
    for (int t = 0; t < 8; ++t) acc[t] = (v8f){};

    const float* Arow = A + (size_t)m * K + 2 * half;

    for (int kc = kBase; kc < kEnd; kc += KC) {
        const int cl = min(KC, kEnd - kc);  // always a multiple of 4 here
        const float* wsrc = W + (size_t)kc * NC;   // chunk is contiguous (row-major)
#if USE_ASYNC_LDS
        for (int i = tid; i < cl * (NC / 4); i += 256) {
            unsigned     ldsOff = (unsigned)(size_t)(void*)&s_w[i * 4];
            const float* gsrc   = wsrc + (size_t)i * 4;
            asm volatile("global_load_async_to_lds_b128 %0, %1, off"
                         :: "v"(ldsOff), "v"(gsrc) : "memory");
        }
        asm volatile("s_wait_asynccnt 0x0" ::: "memory");
        __syncthreads();
#else
        for (int i = tid; i < cl * (NC / 4); i += 256)
            ((float4*)s_w)[i] = ((const float4*)wsrc)[i];
        __syncthreads();
#endif
        for (int k = 0; k < cl; k += 4) {
            v2f a = *reinterpret_cast<const v2f*>(Arow + kc + k);   // 8B-aligned (K even)
            const float* b0 = &s_w[(k + 2 * half) * NC + r];
            const float* b1 = b0 + NC;                              // next K row (+512B)
            #pragma unroll
            for (int t = 0; t < 8; ++t) {
                v2f b; b.x = b0[t * 16]; b.y = b1[t * 16];
                acc[t] = __builtin_amdgcn_wmma_f32_16x16x4_f32(
                    false, a, false, b, (short)0, acc[t], false, false);
            }
        }
        __syncthreads();
    }

    #pragma unroll
    for (int t = 0; t < 8; ++t) {
        int n = t * 16 + r;
        float bv = bias ? bias[n] : 0.0f;
        #pragma unroll
        for (int v = 0; v < 8; ++v) {
            int row = tm * 16 + v + 8 * half;   // C layout: VGPR v -> M=v (+8 for hi half)
            Cw[(size_t)row * NC + n] = acc[t][v] + bv;
        }
    }
}

// ---------------------------------------------------------------------------
// 4) Deterministic reduction of lincomb K-split partials (+ bias).
// ---------------------------------------------------------------------------
__global__ void reduce_partials(const float* __restrict__ part, const float* __restrict__ bias,
                                float* __restrict__ g, int chunks) {
    int i = blockIdx.x * blockDim.x + threadIdx.x;      // 0 .. BGR*HID-1
    if (i >= BGR * HID) return;
    float acc = bias[i & (HID - 1)];
    for (int c = 0; c < chunks; ++c) acc += part[(size_t)c * BGR * HID + i];
    g[i] = acc;
}

// ---------------------------------------------------------------------------
// 5) Aggregation: out[n,h] = bias[h] + dinv[n]^2*xw[n,h] + sum_k norm_k*xw[src_k,h]
//    One 128-thread block per node; src/norm staged in LDS; per-k the lanes
//    read one contiguous 512B xw row (per-graph 100KB working set -> L2 hits).
// ---------------------------------------------------------------------------
__global__ void aggregate_kernel(const float* __restrict__ xw, const float* __restrict__ nrm,
                                 const int* __restrict__ src, const float* __restrict__ dinv,
                                 const float* __restrict__ bias, float* __restrict__ out,
                                 int leaky) {
    __shared__ int   s_src[DEG];
    __shared__ float s_nrm[DEG];
    int n = blockIdx.x;
    int h = threadIdx.x;
    if (h < DEG) {
        s_src[h] = src[(size_t)n * DEG + h];
        s_nrm[h] = nrm[(size_t)n * DEG + h];
    }
    __syncthreads();

    float di  = dinv[n];
    float acc = di * di * xw[(size_t)n * HID + h] + bias[h];
    #pragma unroll 4
    for (int k = 0; k < DEG; ++k)
        acc += s_nrm[k] * xw[(size_t)s_src[k] * HID + h];
    if (leaky) acc = acc > 0.0f ? acc : 0.01f * acc;
    out[(size_t)n * HID + h] = acc;
}

// ---------------------------------------------------------------------------
// 6) Tiny FC layers (128xK @ KxNc): one thread per output element.
// ---------------------------------------------------------------------------
__global__ void fc_kernel(const float* __restrict__ in, const float* __restrict__ W,
                          const float* __restrict__ bias, float* __restrict__ out,
                          int Bn, int K, int Nc, int leaky) {
    int idx = blockIdx.x * blockDim.x + threadIdx.x;
    if (idx >= Bn * Nc) return;
    int b = idx / Nc, j = idx - b * Nc;
    float acc = bias[j];
    for (int k = 0; k < K; ++k) acc += in[(size_t)b * K + k] * W[(size_t)k * Nc + j];
    if (leaky) acc = acc > 0.0f ? acc : 0.01f * acc;
    out[idx] = acc;
}

// ---------------------------------------------------------------------------
extern "C" void kernel_launch(void* const* d_in, const int* in_sizes, int n_in,
                              void* d_out, int out_size, void* d_ws, size_t ws_size,
                              hipStream_t stream) {
    (void)in_sizes; (void)n_in; (void)out_size; (void)ws_size;

    const float* x      = (const float*)d_in[0];
    const int*   ei     = (const int*)  d_in[1];
    const float* ew     = (const float*)d_in[2];
    const float* W0     = (const float*)d_in[3];
    const float* b0     = (const float*)d_in[4];
    const float* W1     = (const float*)d_in[5];
    const float* b1     = (const float*)d_in[6];
    const float* W2     = (const float*)d_in[7];
    const float* b2     = (const float*)d_in[8];
    const float* lincW  = (const float*)d_in[9];
    const float* lincB  = (const float*)d_in[10];
    const float* fc1W   = (const float*)d_in[11];
    const float* fc1b   = (const float*)d_in[12];
    const float* fc2W   = (const float*)d_in[13];
    const float* fc2b   = (const float*)d_in[14];
    const float* fc3W   = (const float*)d_in[15];
    const float* fc3b   = (const float*)d_in[16];

    const int* src = ei;
    const int* dst = ei + EE;

    // Workspace (floats): dinv | norm | bufX | bufH | g | t1 | t2 | lincomb partials
    float* ws    = (float*)d_ws;
    float* dinv  = ws;
    float* nrm   = dinv + NN;
    float* bufX  = nrm  + EE;
    float* bufH  = bufX + (size_t)NN * HID;
    float* g     = bufH + (size_t)NN * HID;
    float* t1    = g    + BGR * HID;
    float* t2    = t1   + BGR * 64;
    float* part  = t2   + BGR * 32;          // LIN_CHUNKS * BGR * HID floats

    deg_dinv_kernel<<<NN / 8, 256, 0, stream>>>(ew, dinv);
    norm_kernel<<<(int)(EE / 256), 256, 0, stream>>>(src, dst, ew, dinv, nrm);

    // Layer 0: xw = x @ W0 ; aggregate + b0 ; leaky
    gemm_strip_wmma<<<dim3(NN / 128, 1), 256, 0, stream>>>(x, W0, bufX, nullptr, FIN, FIN, 0);
    aggregate_kernel<<<NN, HID, 0, stream>>>(bufX, nrm, src, dinv, b0, bufH, 1);

    // Layer 1
    gemm_strip_wmma<<<dim3(NN / 128, 1), 256, 0, stream>>>(bufH, W1, bufX, nullptr, HID, HID, 0);
    aggregate_kernel<<<NN, HID, 0, stream>>>(bufX, nrm, src, dinv, b1, bufH, 1);

    // Layer 2 (no activation)
    gemm_strip_wmma<<<dim3(NN / 128, 1), 256, 0, stream>>>(bufH, W2, bufX, nullptr, HID, HID, 0);
    aggregate_kernel<<<NN, HID, 0, stream>>>(bufX, nrm, src, dinv, b2, bufH, 0);

    // lincomb: [128, 25600] @ [25600, 128], deterministic K-split into 100 partials
    gemm_strip_wmma<<<dim3(1, LIN_CHUNKS), 256, 0, stream>>>(
        bufH, lincW, part, nullptr, NSN * HID, LIN_KSPAN, (size_t)BGR * HID);
    reduce_partials<<<(BGR * HID + 255) / 256, 256, 0, stream>>>(part, lincB, g, LIN_CHUNKS);

    // MLP head
    fc_kernel<<<(BGR * 64 + 255) / 256, 256, 0, stream>>>(g,  fc1W, fc1b, t1, BGR, HID, 64, 1);
    fc_kernel<<<(BGR * 32 + 255) / 256, 256, 0, stream>>>(t1, fc2W, fc2b, t2, BGR, 64,  32, 1);
    fc_kernel<<<(BGR * 1  + 255) / 256, 256, 0, stream>>>(t2, fc3W, fc3b, (float*)d_out, BGR, 32, 1, 0);
}